// FocalLoss_32341103739147
// MI455X (gfx1250) — compile-verified
//
#include <hip/hip_runtime.h>
#include <hip/hip_bf16.h>
#include <math.h>

typedef __attribute__((ext_vector_type(4))) float f4;
typedef __attribute__((ext_vector_type(2))) float v2f;
typedef __attribute__((ext_vector_type(8))) float v8f;

#define ROW_W 512
#define ROWS_PER_BLOCK 8
#define ALPHA_C 0.25f

__device__ __forceinline__ float wave_sum(float v) {
#pragma unroll
  for (int off = 16; off > 0; off >>= 1) v += __shfl_xor(v, off, 32);
  return v;
}

__device__ __forceinline__ float wave_max(float v) {
#pragma unroll
  for (int off = 16; off > 0; off >>= 1) v = fmaxf(v, __shfl_xor(v, off, 32));
  return v;
}

// One wave32 per row of 512 floats. Each lane: 4 x float4 (b128, non-temporal).
// Block (8 waves) = 8 consecutive rows of the same (b,c); emits one partial sum
// of row-CE values. Fully deterministic (no atomics).
__global__ __launch_bounds__(256) void focal_row_kernel(
    const float* __restrict__ pred, const float* __restrict__ target,
    float* __restrict__ part) {
  const int wave = threadIdx.x >> 5;
  const int lane = threadIdx.x & 31;
  const long long row = (long long)blockIdx.x * ROWS_PER_BLOCK + wave;
  const float* __restrict__ prow = pred + row * ROW_W;
  const float* __restrict__ trow = target + row * ROW_W;

  f4 p[4];
  float lmax = -INFINITY;
  float s_t = 0.0f;   // sum of target
  float s_tp = 0.0f;  // sum of target * pred

#pragma unroll
  for (int j = 0; j < 4; ++j) {
    // chunk j: 128 consecutive floats across the wave, 16B per lane
    const f4 pp = __builtin_nontemporal_load(
        (const f4*)(prow + j * 128) + lane);
    const f4 tt = __builtin_nontemporal_load(
        (const f4*)(trow + j * 128) + lane);
    p[j] = pp;
    lmax = fmaxf(lmax, fmaxf(fmaxf(pp.x, pp.y), fmaxf(pp.z, pp.w)));
    s_t += (tt.x + tt.y) + (tt.z + tt.w);
    s_tp += pp.x * tt.x + pp.y * tt.y + pp.z * tt.z + pp.w * tt.w;
  }

  const float M = wave_max(lmax);

  float s_e = 0.0f;
#pragma unroll
  for (int j = 0; j < 4; ++j) {
    s_e += (__expf(p[j].x - M) + __expf(p[j].y - M)) +
           (__expf(p[j].z - M) + __expf(p[j].w - M));
  }
  s_e = wave_sum(s_e);
  s_t = wave_sum(s_t);
  s_tp = wave_sum(s_tp);

  __shared__ float sce[ROWS_PER_BLOCK];
  if (lane == 0) {
    // ce_row = -(sum t*logp) = (M + log(sum exp(p-M))) * sum(t) - sum(t*p)
    sce[wave] = (M + __logf(s_e)) * s_t - s_tp;
  }
  __syncthreads();
  if (threadIdx.x == 0) {
    float s = 0.0f;
#pragma unroll
    for (int i = 0; i < ROWS_PER_BLOCK; ++i) s += sce[i];
    part[blockIdx.x] = s;
  }
}

// Single wave32: fold 64 partials per (b,c) -> ce[b,c] -> focal, then finish
// the 32-lane reduction on the matrix pipe with V_WMMA_F32_16X16X4_F32
// (B = ones => every D row = sum over K of that A row).
__global__ __launch_bounds__(32) void focal_final_kernel(
    const float* __restrict__ part, float* __restrict__ out,
    int nbc, int per_bc, float inv_h, float inv_b) {
  const int lane = threadIdx.x;
  float fsum = 0.0f;
  for (int bc = lane; bc < nbc; bc += 32) {
    const float* __restrict__ p = part + (long long)bc * per_bc;
    float s = 0.0f;
    for (int j = 0; j < per_bc; ++j) s += p[j];
    const float ce = s * inv_h;          // mean over H
    const float pt = __expf(-ce);
    const float om = 1.0f - pt;
    fsum += ALPHA_C * om * om * ce;      // focal[b,c]
  }

  // A-matrix 16x4 f32 layout: lanes 0-15 -> (M=lane, K=0/1), lanes 16-31 ->
  // (M=lane-16, K=2/3). Put the lane partial in the first slot, 0 in the
  // second; B = all-ones 4x16 => D[m][n] = fsum[m] + fsum[m+16].
  v2f A; A.x = fsum; A.y = 0.0f;
  v2f Bm; Bm.x = 1.0f; Bm.y = 1.0f;
  v8f C = {};
  v8f D = __builtin_amdgcn_wmma_f32_16x16x4_f32(
      /*neg_a=*/false, A, /*neg_b=*/false, Bm,
      /*c_mod=*/(short)0, C, /*reuse_a=*/false, /*reuse_b=*/false);

  // Lane 0 holds (N=0, M=0..7) in D[0..7]; lane 16 holds (N=0, M=8..15).
  float r = ((D[0] + D[1]) + (D[2] + D[3])) + ((D[4] + D[5]) + (D[6] + D[7]));
  r += __shfl_xor(r, 16, 32);
  if (lane == 0) out[0] = r * inv_b;     // mean over batch of sum over classes
}

extern "C" void kernel_launch(void* const* d_in, const int* in_sizes, int n_in,
                              void* d_out, int out_size, void* d_ws, size_t ws_size,
                              hipStream_t stream) {
  const float* pred = (const float*)d_in[0];
  const float* target = (const float*)d_in[1];
  float* out = (float*)d_out;
  float* part = (float*)d_ws;

  const long long n = in_sizes[0];            // 8*21*512*512
  const int rows = (int)(n / ROW_W);          // 86016 = B*C*H
  const int blocks = rows / ROWS_PER_BLOCK;   // 10752 partials
  const int H = 512;
  const int per_bc = H / ROWS_PER_BLOCK;      // 64 partials per (b,c)
  const int nbc = blocks / per_bc;            // 168 = B*C

  focal_row_kernel<<<blocks, 256, 0, stream>>>(pred, target, part);
  focal_final_kernel<<<1, 32, 0, stream>>>(part, out, nbc, per_bc,
                                           1.0f / (float)H, 1.0f / 8.0f);
}